// SingleScaleDynamics_4672924418115
// MI455X (gfx1250) — compile-verified
//
#include <hip/hip_runtime.h>
#include <hip/hip_bf16.h>
#include <stddef.h>

typedef __attribute__((ext_vector_type(2))) float v2f;
typedef __attribute__((ext_vector_type(4))) float v4f;
typedef __attribute__((ext_vector_type(8))) float v8f;

#define HDIM   64            // hidden size (reference H)
#define NGATE  256           // packed gate outputs: r(64) | u(64) | n_i(64) | n_h(64)
#define NKT    16            // real K tiles of 4   (64/4)
#define NKT_B  17            // K tiles + 1 bias tile (A row of ones x B row of biases)
#define NNT    16            // N tiles of 16  (256/16)
#define WPACK_ELEMS (NNT * NKT_B * 64)   // 17408 floats, 68KB
#define ZSTRIDE 66           // padded LDS row stride (floats) to spread banks

// ---------------------------------------------------------------------------
// Pack kernel: build combined weights directly in WMMA B-fragment layout.
//   B-fragment (4x16 f32, 2 VGPRs): VGPR j, lane l -> K = j + 2*(l>>4),
//   N = l&15. Stored per fragment as 64 floats, element index = l*2 + j,
//   so the main kernel reads one aligned ds_load_b64 per lane per fragment.
// Gate column n in [0,256):
//   n <  128 : W_ih[n] + W_hh[n]           (r and u gates pre-summed)
//   n <  192 : W_ih[n]                     (i_n)
//   else     : W_hh[n-64]                  (h_n)
// Fragment kt==16 is the bias tile: K-row 0 carries the packed bias for
// column n, rows 1..3 are zero.  Paired with a constant A fragment of
// [1,0,0,0] this folds the bias add into the GEMM.
// ---------------------------------------------------------------------------
__global__ void gru_pack(const float* __restrict__ Wih, const float* __restrict__ Whh,
                         const float* __restrict__ bih, const float* __restrict__ bhh,
                         float* __restrict__ wpack) {
    int tid = blockIdx.x * blockDim.x + threadIdx.x;
    if (tid >= WPACK_ELEMS) return;
    int f  = tid >> 6;          // fragment id = nt*17 + kt
    int e  = tid & 63;
    int l  = e >> 1;            // lane
    int j  = e & 1;             // vgpr within pair
    int nt = f / NKT_B;
    int kt = f - nt * NKT_B;
    int n  = nt * 16 + (l & 15);
    float v;
    if (kt < NKT) {
        int k = kt * 4 + j + 2 * (l >> 4);
        if (n < 128)      v = Wih[n * HDIM + k] + Whh[n * HDIM + k];
        else if (n < 192) v = Wih[n * HDIM + k];
        else              v = Whh[(n - 64) * HDIM + k];
    } else {
        int krel = j + 2 * (l >> 4);     // K row within bias tile
        if (krel == 0) {
            if (n < 128)      v = bih[n] + bhh[n];
            else if (n < 192) v = bih[n];
            else              v = bhh[n - 64];
        } else {
            v = 0.0f;
        }
    }
    wpack[tid] = v;
}

// ---------------------------------------------------------------------------
// Main kernel: one wave32 per block, 16 rows of B per wave, T serial steps.
// Per step: 272 x v_wmma_f32_16x16x4_f32 (z[16x64] @ Wcat[64x256] + bias).
// Accumulators start as inline-0 C operands; bias comes from the 17th k-tile.
// ---------------------------------------------------------------------------
__global__ __launch_bounds__(32, 1)
void gru_run(const float* __restrict__ z_init,
             const float* __restrict__ wpack,
             float* __restrict__ out, int T) {
    __shared__ __align__(16) float sW[WPACK_ELEMS];   // 68 KB swizzled weights+bias
    __shared__ __align__(16) float sZ[16 * ZSTRIDE];  // z tile, padded rows

    const int lane = threadIdx.x;      // 0..31
    const int m    = lane & 15;        // column-in-tile / M-in-A
    const int hh   = lane >> 4;        // lane half
    const int row_base = blockIdx.x * 16;

    // Cooperative load of packed weights into LDS (b128 chunks; 17408 % 128 == 0).
    for (int i = lane * 4; i < WPACK_ELEMS; i += 32 * 4)
        *(v4f*)&sW[i] = *(const v4f*)&wpack[i];

    // Constant A fragment for the bias tile: value 1.0 at K-row 0 only.
    // Lane l, VGPR j -> K-row = j + 2*hh, so only lanes 0..15, .x carry 1.0.
    v2f afb;
    afb.x = (hh == 0) ? 1.0f : 0.0f;
    afb.y = 0.0f;

    // Load z_init into C-layout registers (VGPR v, lane: M=v+8*hh, N=m)
    // and mirror into the LDS z buffer (A-fragment source).
    float zc[4][8];
#pragma unroll
    for (int c = 0; c < 4; ++c)
#pragma unroll
        for (int v = 0; v < 8; ++v) {
            int row = v + 8 * hh;
            float z0 = z_init[(size_t)(row_base + row) * HDIM + c * 16 + m];
            zc[c][v] = z0;
            sZ[row * ZSTRIDE + c * 16 + m] = z0;
        }
    __syncthreads();   // single-wave WG: lowers to S_NOP, acts as LDS fence

    for (int t = 0; t < T; ++t) {
        // A fragments: lane l, VGPR j -> z[m, kt*4 + j + 2*hh]  (ds_load_b64)
        v2f af[NKT];
#pragma unroll
        for (int kt = 0; kt < NKT; ++kt)
            af[kt] = *(const v2f*)&sZ[m * ZSTRIDE + kt * 4 + 2 * hh];

        // Zero accumulators -> folded into inline-0 C operand of first WMMA.
        v8f acc[NNT];
#pragma unroll
        for (int nt = 0; nt < NNT; ++nt)
#pragma unroll
            for (int i = 0; i < 8; ++i) acc[nt][i] = 0.0f;

        // GEMM: kt outer so consecutive WMMAs hit independent accumulators
        // and share the A operand. kt==16 is the bias tile.
#pragma unroll
        for (int kt = 0; kt < NKT_B; ++kt) {
            v2f a = (kt < NKT) ? af[kt] : afb;
#pragma unroll
            for (int nt = 0; nt < NNT; ++nt) {
                v2f bf = *(const v2f*)&sW[(nt * NKT_B + kt) * 64 + lane * 2];
                acc[nt] = __builtin_amdgcn_wmma_f32_16x16x4_f32(
                    /*neg_a=*/false, a,
                    /*neg_b=*/false, bf,
                    /*c_mod=*/(short)0, acc[nt],
                    /*reuse_a=*/false, /*reuse_b=*/false);
            }
        }

        __syncthreads();   // all A-fragment reads done before sZ is rewritten

        // Gating. acc[0..3]=r, acc[4..7]=u, acc[8..11]=i_n, acc[12..15]=h_n.
        // Biases already included via the bias k-tile.
#pragma unroll
        for (int c = 0; c < 4; ++c) {
#pragma unroll
            for (int v = 0; v < 8; ++v) {
                float rv = 1.0f / (1.0f + __expf(-acc[c][v]));
                float uv = 1.0f / (1.0f + __expf(-acc[4 + c][v]));
                float nv = tanhf(acc[8 + c][v] + rv * acc[12 + c][v]);
                float zn = (1.0f - uv) * nv + uv * zc[c][v];
                zc[c][v] = zn;
                int row = v + 8 * hh;
                sZ[row * ZSTRIDE + c * 16 + m] = zn;
                out[((size_t)(row_base + row) * T + t) * HDIM + c * 16 + m] = zn;
            }
        }
        __syncthreads();   // z updates visible before next step's A loads
    }
}

// ---------------------------------------------------------------------------
// Harness entry point.
// Inputs: z_init[B,H], W_ih[3H,H], W_hh[3H,H], b_ih[3H], b_hh[3H], num_steps.
// Output: [B, T, H] float32.
// ---------------------------------------------------------------------------
extern "C" void kernel_launch(void* const* d_in, const int* in_sizes, int n_in,
                              void* d_out, int out_size, void* d_ws, size_t ws_size,
                              hipStream_t stream) {
    const float* z_init = (const float*)d_in[0];
    const float* Wih    = (const float*)d_in[1];
    const float* Whh    = (const float*)d_in[2];
    const float* bih    = (const float*)d_in[3];
    const float* bhh    = (const float*)d_in[4];

    const int H  = in_sizes[3] / 3;        // 64
    const int Bn = in_sizes[0] / H;        // 2048
    const int T  = out_size / (Bn * H);    // 64 (num_steps lives on device)

    float* wpack = (float*)d_ws;

    gru_pack<<<(WPACK_ELEMS + 255) / 256, 256, 0, stream>>>(
        Wih, Whh, bih, bhh, wpack);

    gru_run<<<dim3(Bn / 16), dim3(32), 0, stream>>>(
        z_init, wpack, (float*)d_out, T);
}